// AscendRejectionSampler_16372415332758
// MI455X (gfx1250) — compile-verified
//
#include <hip/hip_runtime.h>
#include <cstdint>
#include <cstddef>

// ---------------------------------------------------------------------------
// Rejection sampler for speculative decoding (MI455X / gfx1250).
// Memory-bound streaming workload: no GEMM -> no WMMA. Uses gfx1250 async
// global->LDS DMA (ASYNCcnt path, CDNA5 ISA) to cache each
// 125KB logits row in LDS so target_logits is read from HBM exactly once.
// ---------------------------------------------------------------------------

#define RS_V     32000          // vocab size (compile-time for static LDS)
#define RS_BLK   512            // 16 wave32 waves per workgroup
#define RS_NW    (RS_BLK / 32)
#define NEG_HUGE (-3.402823466e38f)

#if __has_builtin(__builtin_amdgcn_global_load_async_to_lds_b128)
#define RS_HAVE_ASYNC 1
#endif

// The async-LDS builtin takes pointers to 16-byte int vectors:
//   param0: int __vector(4) AS(1)*   (global source)
//   param1: int __vector(4) AS(3)*   (LDS destination)
typedef int rs_v4i __attribute__((vector_size(16)));
typedef __attribute__((address_space(1))) rs_v4i* rs_gv4p;
typedef __attribute__((address_space(3))) rs_v4i* rs_lv4p;

// ---- wave32 reductions ----------------------------------------------------
__device__ __forceinline__ float rs_wave_sum(float v) {
    #pragma unroll
    for (int o = 16; o > 0; o >>= 1) v += __shfl_xor(v, o, 32);
    return v;
}

__device__ __forceinline__ void rs_wave_argmax(float& v, int& i) {
    #pragma unroll
    for (int o = 16; o > 0; o >>= 1) {
        float ov = __shfl_xor(v, o, 32);
        int   oi = __shfl_xor(i, o, 32);
        if (ov > v || (ov == v && oi < i)) { v = ov; i = oi; }  // first-occurrence tie
    }
}

// ---------------------------------------------------------------------------
// Kernel 1: one workgroup per flat draft token.
// Produces per-token accept flag + emitted token into workspace.
// ---------------------------------------------------------------------------
__global__ __launch_bounds__(RS_BLK)
void rs_token_kernel(const float* __restrict__ draft_probs,
                     const float* __restrict__ target_logits,
                     const float* __restrict__ q,
                     const float* __restrict__ uniform_probs,
                     const float* __restrict__ temperature,
                     const int*   __restrict__ draft_token_ids,
                     const int*   __restrict__ cu,
                     int B, int V,
                     int* __restrict__ ws_accept,
                     int* __restrict__ ws_token)
{
    __shared__ float s_row[RS_V];      // cached logits row (125 KB of 320 KB LDS)
    __shared__ float s_sv[RS_NW];
    __shared__ int   s_si[RS_NW];

    const int tok  = blockIdx.x;
    const int tid  = threadIdx.x;
    const int lane = tid & 31;
    const int w    = tid >> 5;
    const int V4   = V >> 2;           // V is a multiple of 4

    const float4* trow = (const float4*)(target_logits + (size_t)tok * V);

    // ---- stage the logits row into LDS ------------------------------------
#if defined(RS_HAVE_ASYNC)
    for (int idx = tid; idx < V4; idx += RS_BLK) {
        __builtin_amdgcn_global_load_async_to_lds_b128(
            (rs_gv4p)(trow + idx),
            (rs_lv4p)(&s_row[idx * 4]),
            /*offset=*/0, /*cpol=*/0);
    }
#if __has_builtin(__builtin_amdgcn_s_wait_asynccnt)
    __builtin_amdgcn_s_wait_asynccnt(0);
#else
    asm volatile("s_wait_asynccnt 0" ::: "memory");
#endif
#else
    for (int idx = tid; idx < V4; idx += RS_BLK)
        ((float4*)s_row)[idx] = trow[idx];
#endif
    __syncthreads();

    // ---- Pass A: row max (softmax shift) + argmax (greedy target token) ---
    float bv = NEG_HUGE;
    int   bi = 0x7fffffff;
    for (int idx = tid; idx < V4; idx += RS_BLK) {
        float4 x = ((const float4*)s_row)[idx];
        int base = idx * 4;
        if (x.x > bv) { bv = x.x; bi = base + 0; }
        if (x.y > bv) { bv = x.y; bi = base + 1; }
        if (x.z > bv) { bv = x.z; bi = base + 2; }
        if (x.w > bv) { bv = x.w; bi = base + 3; }
    }
    rs_wave_argmax(bv, bi);
    if (lane == 0) { s_sv[w] = bv; s_si[w] = bi; }
    __syncthreads();
    if (w == 0) {
        float v = (lane < RS_NW) ? s_sv[lane] : NEG_HUGE;
        int   i = (lane < RS_NW) ? s_si[lane] : 0x7fffffff;
        rs_wave_argmax(v, i);
        if (lane == 0) { s_sv[0] = v; s_si[0] = i; }
    }
    __syncthreads();
    const float M = s_sv[0];
    const int target_argmax = s_si[0];
    __syncthreads();   // release scratch before pass B writes

    // ---- Pass B: softmax denominator S = sum exp(x - M) -------------------
    float acc = 0.0f;
    for (int idx = tid; idx < V4; idx += RS_BLK) {
        float4 x = ((const float4*)s_row)[idx];
        acc += __expf(x.x - M) + __expf(x.y - M) + __expf(x.z - M) + __expf(x.w - M);
    }
    acc = rs_wave_sum(acc);
    if (lane == 0) s_sv[w] = acc;
    __syncthreads();
    if (w == 0) {
        float v = (lane < RS_NW) ? s_sv[lane] : 0.0f;
        v = rs_wave_sum(v);
        if (lane == 0) s_sv[0] = v;
    }
    __syncthreads();
    const float S    = s_sv[0];
    const float invS = 1.0f / S;
    __syncthreads();   // release scratch before pass C writes

    // ---- map token -> request (searchsorted right), uniform across block --
    int req = B - 1;
    { int lo = 0, hi = B - 1;
      while (lo <= hi) { int mid = (lo + hi) >> 1;
                         if (cu[mid] > tok) { req = mid; hi = mid - 1; }
                         else lo = mid + 1; } }
    const bool g  = (temperature[req] == -1.0f);   // greedy request
    const int  dt = draft_token_ids[tok];

    // ---- Pass C (non-greedy only): recovered = argmax (t - d)+ / q --------
    int recovered = 0;
    if (!g) {
        const float4* drow = (const float4*)(draft_probs + (size_t)tok * V);
        const float4* qrow = (const float4*)(q + (size_t)req * V);
        float rv = NEG_HUGE;
        int   ri = 0x7fffffff;
        for (int idx = tid; idx < V4; idx += RS_BLK) {
            float4 d  = drow[idx];
            float4 qq = qrow[idx];
            float4 x  = ((const float4*)s_row)[idx];
            int base = idx * 4;
            float s0 = __fdividef(fmaxf(__expf(x.x - M) * invS - d.x, 0.0f), qq.x);
            float s1 = __fdividef(fmaxf(__expf(x.y - M) * invS - d.y, 0.0f), qq.y);
            float s2 = __fdividef(fmaxf(__expf(x.z - M) * invS - d.z, 0.0f), qq.z);
            float s3 = __fdividef(fmaxf(__expf(x.w - M) * invS - d.w, 0.0f), qq.w);
            if (s0 > rv) { rv = s0; ri = base + 0; }
            if (s1 > rv) { rv = s1; ri = base + 1; }
            if (s2 > rv) { rv = s2; ri = base + 2; }
            if (s3 > rv) { rv = s3; ri = base + 3; }
        }
        rs_wave_argmax(rv, ri);
        if (lane == 0) { s_sv[w] = rv; s_si[w] = ri; }
        __syncthreads();
        if (w == 0) {
            float v = (lane < RS_NW) ? s_sv[lane] : NEG_HUGE;
            int   i = (lane < RS_NW) ? s_si[lane] : 0x7fffffff;
            rs_wave_argmax(v, i);
            if (lane == 0) { s_sv[0] = v; s_si[0] = i; }
        }
        __syncthreads();
        recovered = s_si[0];
    }

    // ---- decision ----------------------------------------------------------
    if (tid == 0) {
        const float p_draft = draft_probs[(size_t)tok * V + dt];
        const float p_tgt   = __expf(s_row[dt] - M) * invS;
        const float u       = uniform_probs[tok];
        const bool accept_r = (p_draft > 0.0f) && (p_tgt >= u * p_draft);
        const bool accept_g = (dt == target_argmax);
        const bool accept   = g ? accept_g : accept_r;
        const int  token    = g ? target_argmax : (accept_r ? dt : recovered);
        ws_accept[tok] = accept ? 1 : 0;
        ws_token[tok]  = token;
    }
}

// ---------------------------------------------------------------------------
// Kernel 2: per-request first-rejection scan + bonus token.
// ---------------------------------------------------------------------------
__global__ void rs_finalize_kernel(const int* __restrict__ ws_accept,
                                   const int* __restrict__ ws_token,
                                   const int* __restrict__ cu,
                                   const int* __restrict__ bonus,
                                   int B, int stride,      // stride = max_spec_len+1
                                   int* __restrict__ out)
{
    int r = blockIdx.x * blockDim.x + threadIdx.x;
    if (r >= B) return;
    const int start = (r > 0) ? cu[r - 1] : 0;
    const int end   = cu[r];
    const int nd    = end - start;
    for (int p = 0; p < stride; ++p) out[r * stride + p] = -1;   // PLACEHOLDER
    int rej = 0;
    for (int p = 0; p < nd; ++p) {
        const int a = ws_accept[start + p];
        const int t = ws_token[start + p];
        if (rej == 0) out[r * stride + p] = t;   // rejected slot still emits its token
        if (!a) rej++;
    }
    if (rej == 0 && nd < stride) out[r * stride + nd] = bonus[r];
}

// ---------------------------------------------------------------------------
extern "C" void kernel_launch(void* const* d_in, const int* in_sizes, int n_in,
                              void* d_out, int out_size, void* d_ws, size_t ws_size,
                              hipStream_t stream)
{
    const float* draft_probs     = (const float*)d_in[0];
    const float* target_logits   = (const float*)d_in[1];
    const float* q               = (const float*)d_in[2];
    const float* uniform_probs   = (const float*)d_in[3];
    const float* temperature     = (const float*)d_in[4];
    const int*   draft_token_ids = (const int*)d_in[5];
    const int*   cu              = (const int*)d_in[6];
    const int*   bonus           = (const int*)d_in[7];
    // d_in[8] = max_spec_len (device scalar) -- derived from out_size instead.

    const int N = in_sizes[5];                 // flattened token count
    const int B = in_sizes[6];                 // number of requests
    const int V = in_sizes[0] / N;             // vocab size
    const int stride = out_size / B;           // max_spec_len + 1

    int* ws_accept = (int*)d_ws;
    int* ws_token  = ws_accept + N;

    if (V > RS_V) return;                      // static LDS sized for reference V

    rs_token_kernel<<<N, RS_BLK, 0, stream>>>(
        draft_probs, target_logits, q, uniform_probs, temperature,
        draft_token_ids, cu, B, V, ws_accept, ws_token);

    rs_finalize_kernel<<<(B + 127) / 128, 128, 0, stream>>>(
        ws_accept, ws_token, cu, bonus, B, stride, (int*)d_out);
}